// MeanDistancer_11510512353607
// MI455X (gfx1250) — compile-verified
//
#include <hip/hip_runtime.h>

// ---------------------------------------------------------------------------
// 5-point |grad| stencil, edge-clamped:  out = 0.25*(|c-up|+|c-down|+|c-left|+|c-right|)
// Input  t: (32, 3, 1024, 1024) f32, only channel 0 used.
// Output  : (32, 1, 1024, 1024) f32.
// Bandwidth bound: ~256 MB total traffic -> ~11us at 23.3 TB/s.
// CDNA5 path: async global->LDS row copies (ASYNCcnt) + wave32 float4 compute.
// NT stores keep the 128MB input working set live in the 192MB L2 so halo
// re-reads by adjacent row-strips hit L2 instead of HBM.
// ---------------------------------------------------------------------------

#define H 1024
#define W 1024
#define TY 16                 // output rows per block
#define NROWS (TY + 2)        // rows resident in LDS (with top/bottom halo)

#if __has_builtin(__builtin_amdgcn_global_load_async_to_lds_b128)
#define ASYNC_LDS 1
#else
#define ASYNC_LDS 0
#endif

typedef __attribute__((__vector_size__(4 * sizeof(int))))   int   vint4;
typedef __attribute__((__vector_size__(4 * sizeof(float)))) float vfloat4;

__device__ __forceinline__ void copy_row16_to_lds(const float* g, float* l) {
#if ASYNC_LDS
    // 16 bytes per lane, global -> LDS via the async data path (no VGPR staging).
    __builtin_amdgcn_global_load_async_to_lds_b128(
        (vint4*)(void*)g, (vint4*)(void*)l, /*imm offset*/0, /*cpol*/0);
#else
    *(float4*)l = *(const float4*)g;   // fallback: sync copy through VGPRs
#endif
}

__device__ __forceinline__ void wait_async_copies() {
#if ASYNC_LDS
#if __has_builtin(__builtin_amdgcn_s_wait_asynccnt)
    __builtin_amdgcn_s_wait_asynccnt(0);
#else
    asm volatile("s_wait_asynccnt 0x0" ::: "memory");
#endif
#endif
}

__global__ __launch_bounds__(256) void mean_dist_stencil(
    const float* __restrict__ t, float* __restrict__ out) {
    __shared__ float tile[NROWS * W];          // 18 * 4KB = 72KB LDS

    const int tid = threadIdx.x;               // 0..255
    const int b   = blockIdx.y;                // batch
    const int y0  = blockIdx.x * TY;           // first output row of this strip
    const int c4  = tid * 4;                   // first column handled (float4)

    const float* __restrict__ xb = t + (size_t)b * 3u * H * W;   // channel 0

    // ---- Stage NROWS full-width rows (edge-clamped) into LDS asynchronously.
#pragma unroll
    for (int r = 0; r < NROWS; ++r) {
        int gy = y0 - 1 + r;
        gy = gy < 0 ? 0 : (gy > H - 1 ? H - 1 : gy);
        copy_row16_to_lds(xb + (size_t)gy * W + c4, &tile[r * W + c4]);
    }
    wait_async_copies();       // this wave's ASYNCcnt -> 0
    __syncthreads();           // all waves' rows visible in LDS

    // ---- Compute TY rows of float4 outputs straight out of LDS.
    const int lcol = (c4 == 0) ? 0 : c4 - 1;            // left-halo column (clamped)
    const int rcol = (c4 + 4 > W - 1) ? W - 1 : c4 + 4; // right-halo column (clamped)

    float* __restrict__ ob = out + ((size_t)b * H + y0) * W + c4;

#pragma unroll 4
    for (int y = 0; y < TY; ++y) {
        const float4 c  = *(const float4*)&tile[(y + 1) * W + c4];
        const float4 up = *(const float4*)&tile[(y    ) * W + c4];
        const float4 dn = *(const float4*)&tile[(y + 2) * W + c4];
        const float  lf = tile[(y + 1) * W + lcol];
        const float  rt = tile[(y + 1) * W + rcol];

        vfloat4 o;
        o[0] = 0.25f * (fabsf(c.x - up.x) + fabsf(c.x - dn.x) +
                        fabsf(c.x - lf)   + fabsf(c.x - c.y));
        o[1] = 0.25f * (fabsf(c.y - up.y) + fabsf(c.y - dn.y) +
                        fabsf(c.y - c.x)  + fabsf(c.y - c.z));
        o[2] = 0.25f * (fabsf(c.z - up.z) + fabsf(c.z - dn.z) +
                        fabsf(c.z - c.y)  + fabsf(c.z - c.w));
        o[3] = 0.25f * (fabsf(c.w - up.w) + fabsf(c.w - dn.w) +
                        fabsf(c.w - c.z)  + fabsf(c.w - rt));

        // Non-temporal global_store_b128: output is never re-read, keep it
        // from evicting the input halo working set out of L2.
        __builtin_nontemporal_store(o, (vfloat4*)&ob[(size_t)y * W]);
    }
}

extern "C" void kernel_launch(void* const* d_in, const int* in_sizes, int n_in,
                              void* d_out, int out_size, void* d_ws, size_t ws_size,
                              hipStream_t stream) {
    const float* t = (const float*)d_in[0];
    float* out = (float*)d_out;

    const int B = in_sizes[0] / (3 * H * W);   // 32 for the reference shapes

    dim3 grid(H / TY, B);                      // (64, 32) blocks
    dim3 block(256);                           // 8 wave32s
    mean_dist_stencil<<<grid, block, 0, stream>>>(t, out);
}